// MultiBoxLoss_16398185136649
// MI455X (gfx1250) — compile-verified
//
#include <hip/hip_runtime.h>
#include <math.h>

#define BLK    256
#define TILE   256
#define KC     21          // classes (reference C=21)
#define MAXN   16          // objects per image (reference N=16)
#define CAP    8960        // LDS capacity for top-k values (>= P=8732)
#define GSLICE 8           // streaming slices per image -> B*GSLICE workgroups

constexpr int TILE_BYTES = TILE * KC * 4;              // 21504, multiple of 16
constexpr int CHUNKS     = TILE_BYTES / 16;            // 1344
constexpr int ITERS      = (CHUNKS + BLK - 1) / BLK;   // 6 async b128 per thread-slot
static_assert(TILE_BYTES % 16 == 0, "tile must be 16B aligned");
static_assert(ITERS == 6, "s_wait_asynccnt immediate below assumes 6");

// ---- CDNA5 async global->LDS DMA (gfx1250), non-temporal streaming ---------
__device__ __forceinline__ void async_ld_b128(unsigned ldsOff, unsigned long long gaddr) {
    // vdst = per-lane LDS byte offset, vaddr = 64-bit global address
    asm volatile("global_load_async_to_lds_b128 %0, %1, off th:TH_LOAD_NT"
                 :: "v"(ldsOff), "v"(gaddr) : "memory");
}
#define S_WAIT_ASYNC_6() asm volatile("s_wait_asynccnt 6" ::: "memory")
#define S_WAIT_ASYNC_0() asm volatile("s_wait_asynccnt 0" ::: "memory")

__device__ __forceinline__ float smooth_l1(float d) {
    float a = fabsf(d);
    return (a < 1.0f) ? 0.5f * d * d : a - 0.5f;
}

// =============================================================================
// Kernel 0: per-image, per-object best prior (argmax over P, tie -> smallest p)
// One workgroup per image; result (16 prior indices) -> workspace.
// =============================================================================
__global__ __launch_bounds__(BLK) void mb_match(
    const float* __restrict__ boxes,    // [B,N,4] xy
    const float* __restrict__ priors,   // [P,4] cxcy
    int* __restrict__ ovPArr,           // [B,MAXN]
    int B, int P, int N)
{
    __shared__ float sBox[MAXN][4];
    __shared__ float sArea[MAXN];
    __shared__ unsigned long long sKey[MAXN];

    const int b   = blockIdx.x;
    const int tid = threadIdx.x;
    const int n   = (N < MAXN) ? N : MAXN;

    if (tid < n) {
        const float* bb = boxes + ((size_t)b * N + tid) * 4;
        sBox[tid][0] = bb[0]; sBox[tid][1] = bb[1];
        sBox[tid][2] = bb[2]; sBox[tid][3] = bb[3];
        sArea[tid] = (bb[2] - bb[0]) * (bb[3] - bb[1]);
        sKey[tid]  = 0ull;
    }
    __syncthreads();

    const float4* pr4 = (const float4*)priors;
    float    bv[MAXN];
    unsigned bp[MAXN];
#pragma unroll
    for (int j = 0; j < MAXN; ++j) { bv[j] = -1.0f; bp[j] = 0u; }
    for (int p = tid; p < P; p += BLK) {
        float4 pc = pr4[p];
        float px1 = pc.x - 0.5f * pc.z, py1 = pc.y - 0.5f * pc.w;
        float px2 = pc.x + 0.5f * pc.z, py2 = pc.y + 0.5f * pc.w;
        float pa  = pc.z * pc.w;
#pragma unroll
        for (int j = 0; j < MAXN; ++j) {
            if (j < n) {
                float ix = fminf(sBox[j][2], px2) - fmaxf(sBox[j][0], px1);
                float iy = fminf(sBox[j][3], py2) - fmaxf(sBox[j][1], py1);
                ix = fmaxf(ix, 0.0f); iy = fmaxf(iy, 0.0f);
                float inter = ix * iy;
                float iou   = inter / (sArea[j] + pa - inter);
                if (iou > bv[j]) { bv[j] = iou; bp[j] = (unsigned)p; }
            }
        }
    }
#pragma unroll
    for (int j = 0; j < MAXN; ++j) {
        if (j < n && bv[j] >= 0.0f) {
            unsigned long long key =
                ((unsigned long long)__float_as_uint(bv[j]) << 32) |
                (unsigned long long)(0xFFFFFFFFu - bp[j]);   // tie -> min p
            atomicMax(&sKey[j], key);
        }
    }
    __syncthreads();
    if (tid == 0) {
        for (int j = 0; j < MAXN; ++j)
            ovPArr[b * MAXN + j] =
                (j < n) ? (int)(0xFFFFFFFFu - (unsigned)(sKey[j] & 0xFFFFFFFFull)) : -1;
    }
}

// =============================================================================
// Kernel 1: streaming matching + loc loss + cross-entropy, grid = B * GSLICE.
// Slice g of image b owns tiles {t : t % GSLICE == g}; every tile base stays
// 16B aligned (TILE*KC*4 % 16 == 0). Async DMA double-buffers score tiles.
// =============================================================================
__global__ __launch_bounds__(BLK) void mb_slice(
    const float* __restrict__ plocs,    // [B,P,4]
    const float* __restrict__ pscores,  // [B,P,KC]
    const float* __restrict__ boxes,    // [B,N,4]
    const int*   __restrict__ labels,   // [B,N]
    const float* __restrict__ priors,   // [P,4]
    const int*   __restrict__ ovPArr,   // [B,MAXN]
    float* __restrict__ confNeg,        // [B,P]
    int*   __restrict__ nPosS,          // [B*GSLICE]
    float* __restrict__ locS,           // [B*GSLICE]
    float* __restrict__ posS,           // [B*GSLICE]
    int B, int P, int N)
{
    __shared__ __align__(16) float sTile[2][TILE * KC];
    __shared__ __align__(16) float sDummy[4];
    __shared__ float sBox[MAXN][4];
    __shared__ float sArea[MAXN];
    __shared__ int   sLab[MAXN];
    __shared__ int   sOvP[MAXN];
    __shared__ float sRedF[BLK];
    __shared__ float sRedG[BLK];
    __shared__ int   sRedI[BLK];

    const int b   = blockIdx.x / GSLICE;
    const int g   = blockIdx.x % GSLICE;
    const int tid = threadIdx.x;
    const int n   = (N < MAXN) ? N : MAXN;

    if (tid < n) {
        const float* bb = boxes + ((size_t)b * N + tid) * 4;
        sBox[tid][0] = bb[0]; sBox[tid][1] = bb[1];
        sBox[tid][2] = bb[2]; sBox[tid][3] = bb[3];
        sArea[tid] = (bb[2] - bb[0]) * (bb[3] - bb[1]);
        sLab[tid]  = labels[(size_t)b * N + tid];
        sOvP[tid]  = ovPArr[b * MAXN + tid];
    }
    __syncthreads();

    const float4* pr4 = (const float4*)priors;
    const float4* pl4 = (const float4*)plocs;
    const float* imgScores = pscores + (size_t)b * P * KC;

    const int numTiles = (P + TILE - 1) / TILE;
    const int numMy    = (numTiles > g) ? (numTiles - g + GSLICE - 1) / GSLICE : 0;

    auto issue_tile = [&](int t, int bufSel) {
        int t0 = t * TILE;
        int tc = (P - t0 < TILE) ? (P - t0) : TILE;
        int tileBytes = ((tc * KC * 4) + 15) & ~15;
        unsigned bufOff = (unsigned)(size_t)(&sTile[bufSel][0]);
        unsigned dumOff = (unsigned)(size_t)(&sDummy[0]);
        unsigned long long gbase =
            (unsigned long long)(const void*)(imgScores + (size_t)t0 * KC);
#pragma unroll
        for (int k = 0; k < ITERS; ++k) {              // fixed 6 issues per wave
            int off = (tid + k * BLK) * 16;
            bool valid = off < tileBytes;
            unsigned l = valid ? (bufOff + (unsigned)off) : dumOff;
            unsigned long long gg = valid ? (gbase + (unsigned long long)off) : gbase;
            async_ld_b128(l, gg);
        }
    };
    auto issue_dummy = [&]() {
        unsigned dumOff = (unsigned)(size_t)(&sDummy[0]);
        unsigned long long gg = (unsigned long long)(const void*)imgScores;
#pragma unroll
        for (int k = 0; k < ITERS; ++k) async_ld_b128(dumOff, gg);
    };

    float locLocal = 0.0f, posLocal = 0.0f;
    int   npLocal  = 0;

    if (numMy > 0) {
        issue_tile(g, 0);
        for (int i = 0; i < numMy; ++i) {
            if (i + 1 < numMy) issue_tile(g + (i + 1) * GSLICE, (i + 1) & 1);
            else               issue_dummy();          // keep async count fixed

            const int t  = g + i * GSLICE;
            const int t0 = t * TILE;
            const int tc = (P - t0 < TILE) ? (P - t0) : TILE;
            const int p  = t0 + tid;
            const bool active = (tid < tc);

            int myLab = 0;
            if (active) {
                float4 pc = pr4[p];
                float px1 = pc.x - 0.5f * pc.z, py1 = pc.y - 0.5f * pc.w;
                float px2 = pc.x + 0.5f * pc.z, py2 = pc.y + 0.5f * pc.w;
                float pa  = pc.z * pc.w;
                int bj = 0; float bvp = -1.0f;
                for (int j = 0; j < n; ++j) {          // argmax axis=0, tie -> min j
                    float ix = fminf(sBox[j][2], px2) - fmaxf(sBox[j][0], px1);
                    float iy = fminf(sBox[j][3], py2) - fmaxf(sBox[j][1], py1);
                    ix = fmaxf(ix, 0.0f); iy = fmaxf(iy, 0.0f);
                    float inter = ix * iy;
                    float iou = inter / (sArea[j] + pa - inter);
                    if (iou > bvp) { bvp = iou; bj = j; }
                }
                for (int j = 0; j < n; ++j)            // forced matches, later j wins
                    if (sOvP[j] == p) { bj = j; bvp = 1.0f; }
                myLab = (bvp < 0.5f) ? 0 : sLab[bj];
                if (myLab != 0) {
                    ++npLocal;
                    float x1 = sBox[bj][0], y1 = sBox[bj][1];
                    float x2 = sBox[bj][2], y2 = sBox[bj][3];
                    float cx = 0.5f * (x1 + x2), cy = 0.5f * (y1 + y2);
                    float w  = x2 - x1,          h  = y2 - y1;
                    float gx = 10.0f * (cx - pc.x) / pc.z;
                    float gy = 10.0f * (cy - pc.y) / pc.w;
                    float gw = 5.0f * logf(w / pc.z);
                    float gh = 5.0f * logf(h / pc.w);
                    float4 q = pl4[(size_t)b * P + p]; // only touched at positives
                    locLocal += smooth_l1(q.x - gx) + smooth_l1(q.y - gy) +
                                smooth_l1(q.z - gw) + smooth_l1(q.w - gh);
                }
            }

            S_WAIT_ASYNC_6();    // <=6 outstanding allowed -> tile t is in LDS
            __syncthreads();

            if (active) {
                const float* s = &sTile[i & 1][tid * KC];
                float m = s[0];
                for (int c = 1; c < KC; ++c) m = fmaxf(m, s[c]);
                float se = 0.0f;
                for (int c = 0; c < KC; ++c) se += expf(s[c] - m);
                float conf = m + logf(se) - s[myLab];  // -log_softmax[label]
                bool pos = (myLab != 0);
                if (pos) posLocal += conf;
                confNeg[(size_t)b * P + p] = pos ? 0.0f : fmaxf(conf, 0.0f);
            }
            __syncthreads();     // buffer (i&1) free before it is re-filled
        }
        S_WAIT_ASYNC_0();        // drain dummy loads
    }

    // ---- deterministic block reductions ----
    sRedF[tid] = locLocal; sRedG[tid] = posLocal; sRedI[tid] = npLocal;
    __syncthreads();
    for (int s = BLK / 2; s > 0; s >>= 1) {
        if (tid < s) {
            sRedF[tid] += sRedF[tid + s];
            sRedG[tid] += sRedG[tid + s];
            sRedI[tid] += sRedI[tid + s];
        }
        __syncthreads();
    }
    if (tid == 0) {
        locS[blockIdx.x]  = sRedF[0];
        posS[blockIdx.x]  = sRedG[0];
        nPosS[blockIdx.x] = sRedI[0];
    }
}

// =============================================================================
// Kernel 2: exact top-k sum of negatives per image (k = 3*n_pos).
// Bit-prefix selection of the k-th largest value (valid: all values >= 0).
// =============================================================================
__global__ __launch_bounds__(BLK) void mb_topk(
    const float* __restrict__ confNeg, const int* __restrict__ nPosS,
    float* __restrict__ hardArr, int P)
{
    __shared__ float    sv[CAP];
    __shared__ unsigned sCnt;
    __shared__ float    sRed[BLK];

    const int b = blockIdx.x, tid = threadIdx.x;
    const float* g = confNeg + (size_t)b * P;
    const int lim = (P < CAP) ? P : CAP;
    for (int i = tid; i < lim; i += BLK) sv[i] = g[i];
    __syncthreads();

    int k = 0;
    for (int s = 0; s < GSLICE; ++s) k += nPosS[b * GSLICE + s];
    k *= 3;
    if (k > P) k = P;
    if (k <= 0) { if (tid == 0) hardArr[b] = 0.0f; return; }   // uniform exit

    unsigned prefix = 0;
    for (int bit = 31; bit >= 0; --bit) {
        unsigned cand = prefix | (1u << bit);
        if (tid == 0) sCnt = 0u;
        __syncthreads();
        unsigned local = 0;
        for (int i = tid; i < P; i += BLK) {
            float x = (i < CAP) ? sv[i] : g[i];
            local += (__float_as_uint(x) >= cand) ? 1u : 0u;
        }
        atomicAdd(&sCnt, local);
        __syncthreads();
        if (sCnt >= (unsigned)k) prefix = cand;   // uniform decision
        __syncthreads();
    }

    if (tid == 0) sCnt = 0u;
    __syncthreads();
    unsigned localm = 0; float localsum = 0.0f;
    for (int i = tid; i < P; i += BLK) {
        float x = (i < CAP) ? sv[i] : g[i];
        if (__float_as_uint(x) > prefix) { ++localm; localsum += x; }
    }
    atomicAdd(&sCnt, localm);
    sRed[tid] = localsum;
    __syncthreads();
    for (int s = BLK / 2; s > 0; s >>= 1) {
        if (tid < s) sRed[tid] += sRed[tid + s];
        __syncthreads();
    }
    if (tid == 0) {
        float t = __uint_as_float(prefix);                    // k-th largest value
        hardArr[b] = sRed[0] + (float)(k - (int)sCnt) * t;    // ties contribute t
    }
}

// =============================================================================
// Kernel 3: deterministic final combine.
// =============================================================================
__global__ void mb_final(const int* __restrict__ nPosS,
                         const float* __restrict__ locS,
                         const float* __restrict__ posS,
                         const float* __restrict__ hardArr,
                         int B, float* __restrict__ out)
{
    if (threadIdx.x == 0 && blockIdx.x == 0) {
        int np = 0; float l = 0.0f, p = 0.0f, h = 0.0f;
        for (int i = 0; i < B * GSLICE; ++i) {
            np += nPosS[i]; l += locS[i]; p += posS[i];
        }
        for (int b = 0; b < B; ++b) h += hardArr[b];
        float Nf = (float)np;
        out[0] = (h + p) / Nf + l / (Nf * 4.0f);
    }
}

extern "C" void kernel_launch(void* const* d_in, const int* in_sizes, int n_in,
                              void* d_out, int out_size, void* d_ws, size_t ws_size,
                              hipStream_t stream) {
    const float* plocs   = (const float*)d_in[0];
    const float* pscores = (const float*)d_in[1];
    const float* boxes   = (const float*)d_in[2];
    const int*   labels  = (const int*)d_in[3];
    const float* priors  = (const float*)d_in[4];

    const int P = in_sizes[4] / 4;
    const int B = in_sizes[0] / (4 * P);
    const int N = in_sizes[3] / B;

    float* confNeg = (float*)d_ws;                           // B*P floats
    int*   ovPArr  = (int*)(confNeg + (size_t)B * P);        // B*MAXN ints
    int*   nPosS   = ovPArr + B * MAXN;                      // B*GSLICE ints
    float* locS    = (float*)(nPosS + B * GSLICE);           // B*GSLICE floats
    float* posS    = locS + B * GSLICE;                      // B*GSLICE floats
    float* hardS   = posS + B * GSLICE;                      // B floats

    mb_match<<<B, BLK, 0, stream>>>(boxes, priors, ovPArr, B, P, N);
    mb_slice<<<B * GSLICE, BLK, 0, stream>>>(plocs, pscores, boxes, labels, priors,
                                             ovPArr, confNeg, nPosS, locS, posS,
                                             B, P, N);
    mb_topk<<<B, BLK, 0, stream>>>(confNeg, nPosS, hardS, P);
    mb_final<<<1, 64, 0, stream>>>(nPosS, locS, posS, hardS, B, (float*)d_out);
}